// AttentionPool3D_37658273251382
// MI455X (gfx1250) — compile-verified
//
#include <hip/hip_runtime.h>
#include <hip/hip_bf16.h>

// Problem constants (from reference setup_inputs)
#define BATCH 4
#define CDIM 256
#define SDIM 73728          // 32*48*48
#define QDIM 4
#define OUTDIM 512
#define SCALE 0.0625f       // 1/sqrt(256)

// Flash tiling
#define TS 128                       // s-columns per chunk (LDS tile = 256 x 128 floats)
#define GCH 4                        // chunks processed per block (online softmax)
#define NCG (SDIM / (TS * GCH))      // 144 chunk-groups per batch
#define XS_STRIDE 130                // padded LDS row stride (floats) -> conflict-free WMMA B reads
#define SC_STRIDE 132                // padded stride for local score rows

// Dynamic LDS partition (floats)
#define XS_FLOATS   (CDIM * XS_STRIDE)   // 33280
#define QK_FLOATS   (16 * CDIM)          // 4096
#define SC_FLOATS   (QDIM * SC_STRIDE)   // 528
#define SM_FLOATS   (XS_FLOATS + QK_FLOATS + SC_FLOATS + 12)
#define SM_BYTES    (SM_FLOATS * 4)      // 151,664 B  (2 blocks = 303 KB <= 320 KB/WGP)

typedef __attribute__((ext_vector_type(2))) float v2f;
typedef __attribute__((ext_vector_type(8))) float v8f;

// ---------------------------------------------------------------------------
// Kernel 1: qk[16][256] (rows 4..15 zero). qk[q][ci] = scale * sum_co q[q][co]*Wk[co][ci]
// (the q.bk score term is constant over S and cancels in softmax)
// ---------------------------------------------------------------------------
__global__ __launch_bounds__(256)
void qk_prep_kernel(const float* __restrict__ queries, const float* __restrict__ Wk,
                    float* __restrict__ qk) {
    const int row = blockIdx.x;      // 0..15
    const int c   = threadIdx.x;     // 0..255
    float acc = 0.f;
    if (row < QDIM) {
        #pragma unroll 8
        for (int co = 0; co < CDIM; ++co)
            acc += queries[row * CDIM + co] * Wk[co * CDIM + c];
        acc *= SCALE;
    }
    qk[row * CDIM + c] = acc;
}

// ---------------------------------------------------------------------------
// Kernel 2: fused flash attention-pool. One pass over x.
//   per block: (b, chunk-group of GCH*TS s-columns)
//   per chunk: async-stage x tile -> LDS, WMMA scores, online softmax,
//              rescale accumulators, WMMA exp-weighted pooling.
//   emits partial (m[4], l[4], acc[4][256]) per block.
// ---------------------------------------------------------------------------
__global__ __launch_bounds__(256)
void flash_pool_kernel(const float* __restrict__ x, const float* __restrict__ qk,
                       float* __restrict__ pm, float* __restrict__ pl,
                       float* __restrict__ pacc) {
    extern __shared__ float smem[];
    float* xs     = smem;                       // [256][XS_STRIDE]
    float* qk_lds = xs + XS_FLOATS;             // [16][256]
    float* sc_l   = qk_lds + QK_FLOATS;         // [4][SC_STRIDE]
    float* mstate = sc_l + SC_FLOATS;           // [4]
    float* lstate = mstate + 4;                 // [4]
    float* fstate = lstate + 4;                 // [4]

    const int tid  = threadIdx.x;
    const int wave = tid >> 5;
    const int lane = tid & 31;
    const int half = lane >> 4;                 // K-half of WMMA fragment
    const int n    = lane & 15;                 // B/D column; also A row index
    const int m_eff = (n < QDIM) ? n : (QDIM - 1);   // clamp pad rows (never stored)

    const int cg = blockIdx.x % NCG;
    const int b  = blockIdx.x / NCG;
    const float* xg = x + (size_t)b * CDIM * SDIM;

    // stage qk (16KB) once
    for (int i = tid; i < QK_FLOATS; i += 256) qk_lds[i] = qk[i];
    if (tid < QDIM) { mstate[tid] = -3.4e38f; lstate[tid] = 0.f; }
    __syncthreads();

    const uint32_t xs_base = (uint32_t)(uintptr_t)xs;   // LDS byte offset (low 32 bits)

    v8f accA = {};   // pooled accumulator, c-tile wave*2+0
    v8f accB = {};   // pooled accumulator, c-tile wave*2+1

    for (int g = 0; g < GCH; ++g) {
        const int s0 = (cg * GCH + g) * TS;

        // ---- Phase A: async copy x[b][0:256][s0:s0+128] into LDS (130KB) ----
        // 8192 float4's, 256 lanes, 32 iterations; each lane: 16B global -> 16B LDS
        #pragma unroll 4
        for (int it = 0; it < 32; ++it) {
            const int i  = it * 256 + tid;
            const int c  = i >> 5;              // row 0..255
            const int s4 = i & 31;              // float4 within row
            const float* gp = xg + (size_t)c * SDIM + s0 + s4 * 4;
            const uint32_t la = xs_base + (uint32_t)(c * (XS_STRIDE * 4) + s4 * 16);
            asm volatile("global_load_async_to_lds_b128 %0, %1, off"
                         :: "v"(la), "v"(gp) : "memory");
        }
        asm volatile("s_wait_asynccnt 0" ::: "memory");
        __syncthreads();

        // ---- Phase B: scores for this chunk via V_WMMA_F32_16X16X4_F32 ----
        // wave -> 16-wide s tile; A[m][k]=qk[m][c0+k], B[k][nn]=xs[c0+k][ts+nn]
        {
            const int ts = wave * 16;
            v8f sacc = {};
            #pragma unroll 4
            for (int c0 = 0; c0 < CDIM; c0 += 4) {
                const float* ap = &qk_lds[n * CDIM + c0 + 2 * half];
                v2f A; A.x = ap[0]; A.y = ap[1];
                const float* bp = &xs[(c0 + 2 * half) * XS_STRIDE + ts + n];
                v2f Bf; Bf.x = bp[0]; Bf.y = bp[XS_STRIDE];
                sacc = __builtin_amdgcn_wmma_f32_16x16x4_f32(false, A, false, Bf,
                                                             (short)0, sacc, false, false);
            }
            if (half == 0) {
                #pragma unroll
                for (int r = 0; r < QDIM; ++r)
                    sc_l[r * SC_STRIDE + ts + n] = sacc[r];
            }
        }
        __syncthreads();

        // ---- Phase C: online softmax update (waves 0..3 handle q=wave) ----
        if (wave < QDIM) {
            const int q = wave;
            float v0 = sc_l[q * SC_STRIDE + lane];
            float v1 = sc_l[q * SC_STRIDE + lane + 32];
            float v2 = sc_l[q * SC_STRIDE + lane + 64];
            float v3 = sc_l[q * SC_STRIDE + lane + 96];
            float mc = fmaxf(fmaxf(v0, v1), fmaxf(v2, v3));
            #pragma unroll
            for (int off = 16; off > 0; off >>= 1)
                mc = fmaxf(mc, __shfl_xor(mc, off, 32));
            const float m_old = mstate[q];
            const float m_new = fmaxf(m_old, mc);
            const float f     = __expf(m_old - m_new);
            float e0 = __expf(v0 - m_new), e1 = __expf(v1 - m_new);
            float e2 = __expf(v2 - m_new), e3 = __expf(v3 - m_new);
            sc_l[q * SC_STRIDE + lane]      = e0;
            sc_l[q * SC_STRIDE + lane + 32] = e1;
            sc_l[q * SC_STRIDE + lane + 64] = e2;
            sc_l[q * SC_STRIDE + lane + 96] = e3;
            float ls = e0 + e1 + e2 + e3;
            #pragma unroll
            for (int off = 16; off > 0; off >>= 1)
                ls += __shfl_xor(ls, off, 32);
            if (lane == 0) {
                mstate[q] = m_new;
                lstate[q] = lstate[q] * f + ls;
                fstate[q] = f;
            }
        }
        __syncthreads();

        // ---- Phase D: rescale accumulators, pool this chunk via WMMA ----
        {
            float f0 = fstate[0], f1 = fstate[1], f2 = fstate[2], f3 = fstate[3];
            accA[0] *= f0; accA[1] *= f1; accA[2] *= f2; accA[3] *= f3;
            accB[0] *= f0; accB[1] *= f1; accB[2] *= f2; accB[3] *= f3;

            const float* ap0 = &sc_l[m_eff * SC_STRIDE];     // A row: exp weights
            const int c0a = (wave * 2 + 0) * 16;
            const int c0b = (wave * 2 + 1) * 16;
            const float* bpa = &xs[(c0a + n) * XS_STRIDE];   // B col: x row c0a+n
            const float* bpb = &xs[(c0b + n) * XS_STRIDE];
            #pragma unroll 4
            for (int k0 = 0; k0 < TS; k0 += 4) {
                const int k = k0 + 2 * half;
                v2f A;  A.x  = ap0[k]; A.y  = ap0[k + 1];    // consecutive pair
                v2f Ba; Ba.x = bpa[k]; Ba.y = bpa[k + 1];    // consecutive pair
                v2f Bb; Bb.x = bpb[k]; Bb.y = bpb[k + 1];
                accA = __builtin_amdgcn_wmma_f32_16x16x4_f32(false, A, false, Ba,
                                                             (short)0, accA, false, false);
                accB = __builtin_amdgcn_wmma_f32_16x16x4_f32(false, A, false, Bb,
                                                             (short)0, accB, false, false);
            }
        }
        __syncthreads();   // xs / sc_l reused next chunk
    }

    // ---- emit partials: pm/pl[b][cg][q], pacc[b][cg][q][256] ----
    const int pbase = (b * NCG + cg) * QDIM;
    if (tid < QDIM) { pm[pbase + tid] = mstate[tid]; pl[pbase + tid] = lstate[tid]; }
    if (half == 0) {
        #pragma unroll
        for (int r = 0; r < QDIM; ++r) {
            pacc[(pbase + r) * CDIM + (wave * 2 + 0) * 16 + n] = accA[r];
            pacc[(pbase + r) * CDIM + (wave * 2 + 1) * 16 + n] = accB[r];
        }
    }
}

// ---------------------------------------------------------------------------
// Kernel 3: combine partials -> pooled_norm[bq][c] (deterministic)
// ---------------------------------------------------------------------------
__global__ __launch_bounds__(256)
void flash_reduce_kernel(const float* __restrict__ pm, const float* __restrict__ pl,
                         const float* __restrict__ pacc, float* __restrict__ pooled) {
    __shared__ float red[256];
    const int bq = blockIdx.x;               // 0..15
    const int b = bq >> 2, q = bq & 3;
    const int t = threadIdx.x;

    float lm = -3.4e38f;
    for (int g = t; g < NCG; g += 256) lm = fmaxf(lm, pm[(b * NCG + g) * QDIM + q]);
    red[t] = lm; __syncthreads();
    for (int o = 128; o > 0; o >>= 1) {
        if (t < o) red[t] = fmaxf(red[t], red[t + o]);
        __syncthreads();
    }
    const float M = red[0]; __syncthreads();

    float ls = 0.f;
    for (int g = t; g < NCG; g += 256) {
        const int i = (b * NCG + g) * QDIM + q;
        ls += pl[i] * __expf(pm[i] - M);
    }
    red[t] = ls; __syncthreads();
    for (int o = 128; o > 0; o >>= 1) {
        if (t < o) red[t] += red[t + o];
        __syncthreads();
    }
    const float invL = 1.0f / red[0];

    float acc = 0.f;
    for (int g = 0; g < NCG; ++g) {
        const int i = (b * NCG + g) * QDIM + q;
        acc += pacc[i * CDIM + t] * __expf(pm[i] - M);
    }
    pooled[bq * CDIM + t] = acc * invL;
}

// ---------------------------------------------------------------------------
// Kernel 4: attended[b][q][c'] = bv[c'] + sum_c Wv[c'][c] * pooled[bq][c]
// ---------------------------------------------------------------------------
__global__ __launch_bounds__(256)
void attended_kernel(const float* __restrict__ Wv, const float* __restrict__ bv,
                     const float* __restrict__ pooled, float* __restrict__ attended) {
    const int bq = blockIdx.x;
    const int cp = threadIdx.x;
    float acc = 0.f;
    #pragma unroll 8
    for (int c = 0; c < CDIM; ++c)
        acc += Wv[cp * CDIM + c] * pooled[bq * CDIM + c];
    attended[bq * CDIM + cp] = acc + bv[cp];
}

// ---------------------------------------------------------------------------
// Kernel 5: out[b][o] = bo[o] + sum_j Wo[o][j] * attended_flat[b][j]
// ---------------------------------------------------------------------------
__global__ __launch_bounds__(256)
void out_kernel(const float* __restrict__ Wo, const float* __restrict__ bo,
                const float* __restrict__ attended, float* __restrict__ out) {
    const int idx = blockIdx.x * 256 + threadIdx.x;   // 0..2047
    const int b = idx / OUTDIM;
    const int o = idx % OUTDIM;
    float acc = bo[o];
    #pragma unroll 8
    for (int j = 0; j < QDIM * CDIM; ++j)
        acc += Wo[o * (QDIM * CDIM) + j] * attended[b * (QDIM * CDIM) + j];
    out[idx] = acc;
}

// ---------------------------------------------------------------------------
extern "C" void kernel_launch(void* const* d_in, const int* in_sizes, int n_in,
                              void* d_out, int out_size, void* d_ws, size_t ws_size,
                              hipStream_t stream) {
    const float* x       = (const float*)d_in[0];
    const float* queries = (const float*)d_in[1];
    const float* Wk      = (const float*)d_in[2];
    // d_in[3] = bk : constant over S -> cancels in softmax
    const float* Wv      = (const float*)d_in[4];
    const float* bv      = (const float*)d_in[5];
    const float* Wo      = (const float*)d_in[6];
    const float* bo      = (const float*)d_in[7];
    float* out = (float*)d_out;

    // Workspace layout (floats), total ~2.4 MB
    float* ws       = (float*)d_ws;
    float* qk       = ws;                                  // 4096
    float* pm       = qk + 16 * CDIM;                      // 4*144*4 = 2304
    float* pl       = pm + BATCH * NCG * QDIM;             // 2304
    float* pacc     = pl + BATCH * NCG * QDIM;             // 4*144*4*256 = 589824
    float* pooled   = pacc + BATCH * NCG * QDIM * CDIM;    // 4096
    float* attended = pooled + BATCH * QDIM * CDIM;        // 4096

    // allow >64KB dynamic LDS for the fused kernel (320KB/WGP on CDNA5)
    (void)hipFuncSetAttribute((const void*)flash_pool_kernel,
                              hipFuncAttributeMaxDynamicSharedMemorySize, SM_BYTES);

    qk_prep_kernel<<<16, 256, 0, stream>>>(queries, Wk, qk);
    flash_pool_kernel<<<BATCH * NCG, 256, SM_BYTES, stream>>>(x, qk, pm, pl, pacc);
    flash_reduce_kernel<<<BATCH * QDIM, 256, 0, stream>>>(pm, pl, pacc, pooled);
    attended_kernel<<<BATCH * QDIM, 256, 0, stream>>>(Wv, bv, pooled, attended);
    out_kernel<<<(BATCH * OUTDIM) / 256, 256, 0, stream>>>(Wo, bo, attended, out);
}